// NearestMemoryManager_81346680586266
// MI455X (gfx1250) — compile-verified
//
#include <hip/hip_runtime.h>
#include <hip/hip_bf16.h>

typedef __attribute__((ext_vector_type(2))) float v2f;
typedef __attribute__((ext_vector_type(8))) float v8f;

#define N_POSI 1024
#define N_NEGI 5
#define DDIM   128
#define N_LEMI 20000
#define BATCH  8
#define XROW   (N_POSI + N_NEGI)   /* 1029 */
#define MOM    0.5f

// ---------------------------------------------------------------------------
// similarity[r, n] = sum_k x_pos[r, k] * memory[n, k]
//   r in [0, 8192): r = b*1024 + p,  x_pos row lives at x[(b*1029 + p)*128]
//   n in [0, 20000), k in [0, 128)
// One wave = 16(M) x 64(N) strip: 4 C tiles, A-fragment reused x4.
// Block = 4 waves stacked along M -> 64x64 block tile.
// ---------------------------------------------------------------------------
__global__ __launch_bounds__(128) void sim_gemm_kernel(
    const float* __restrict__ x, const float* __restrict__ mem,
    float* __restrict__ out)
{
    const int lane   = threadIdx.x & 31;
    const int wave   = threadIdx.x >> 5;
    const int m0     = blockIdx.y * 64 + wave * 16;   // M tile origin (always < 8192)
    const int n_base = blockIdx.x * 64;               // N strip origin

    // A fragment (16x4 fp32): lanes 0-15 -> K = k0+0,k0+1 ; lanes 16-31 -> K = k0+2,k0+3
    const int mrow = m0 + (lane & 15);
    const int bb_  = mrow >> 10;
    const int pp_  = mrow & (N_POSI - 1);
    const float* __restrict__ arow = x + (size_t)(bb_ * XROW + pp_) * DDIM;
    const int ka = (lane >> 4) * 2;

    // B fragment (4x16 fp32): rows striped across lanes; per-lane column n, kb in {0,1}
    const int nc = lane & 15;
    const int kb = lane >> 4;
    // Clamp load addresses for the (uniform) invalid edge tiles; guard stores only.
    auto browp = [&](int t) {
        int n = n_base + t * 16 + nc;
        if (n >= N_LEMI) n = N_LEMI - 1;
        return mem + (size_t)n * DDIM;
    };
    const float* __restrict__ brow0 = browp(0);
    const float* __restrict__ brow1 = browp(1);
    const float* __restrict__ brow2 = browp(2);
    const float* __restrict__ brow3 = browp(3);

    v8f acc0 = {}, acc1 = {}, acc2 = {}, acc3 = {};

#pragma unroll 4
    for (int k0 = 0; k0 < DDIM; k0 += 4) {
        v2f a;
        a.x = arow[k0 + ka];
        a.y = arow[k0 + ka + 1];

        v2f b0; b0.x = brow0[k0 + kb]; b0.y = brow0[k0 + kb + 2];
        acc0 = __builtin_amdgcn_wmma_f32_16x16x4_f32(false, a, false, b0,
                                                     (short)0, acc0, false, false);
        v2f b1; b1.x = brow1[k0 + kb]; b1.y = brow1[k0 + kb + 2];
        acc1 = __builtin_amdgcn_wmma_f32_16x16x4_f32(false, a, false, b1,
                                                     (short)0, acc1, false, false);
        v2f b2; b2.x = brow2[k0 + kb]; b2.y = brow2[k0 + kb + 2];
        acc2 = __builtin_amdgcn_wmma_f32_16x16x4_f32(false, a, false, b2,
                                                     (short)0, acc2, false, false);
        v2f b3; b3.x = brow3[k0 + kb]; b3.y = brow3[k0 + kb + 2];
        acc3 = __builtin_amdgcn_wmma_f32_16x16x4_f32(false, a, false, b3,
                                                     (short)0, acc3, false, false);
    }

    // C/D layout: VGPR v, lane halves -> row m0 + (lane>>4)*8 + v, col = n tile + (lane&15)
    const int mstore = m0 + (lane >> 4) * 8;
    auto store_tile = [&](const v8f& c, int t) {
        const int n = n_base + t * 16 + nc;
        if (n < N_LEMI) {
            size_t base = (size_t)mstore * N_LEMI + n;
#pragma unroll
            for (int v = 0; v < 8; ++v)
                out[base + (size_t)v * N_LEMI] = c[v];
        }
    };
    store_tile(acc0, 0);
    store_tile(acc1, 1);
    store_tile(acc2, 2);
    store_tile(acc3, 3);
}

// ---------------------------------------------------------------------------
// noise_similarity[(b*5+n), m] = dot(x[b, 1024+n, :], memory[m, :])
// ---------------------------------------------------------------------------
__global__ void noise_kernel(const float* __restrict__ x,
                             const float* __restrict__ mem,
                             float* __restrict__ outn)
{
    int idx = blockIdx.x * blockDim.x + threadIdx.x;
    if (idx >= BATCH * N_NEGI * N_POSI) return;
    int m  = idx & (N_POSI - 1);
    int bn = idx >> 10;
    int b  = bn / N_NEGI, n = bn % N_NEGI;
    const float4* __restrict__ xr = (const float4*)(x + (size_t)(b * XROW + N_POSI + n) * DDIM);
    const float4* __restrict__ mr = (const float4*)(mem + (size_t)m * DDIM);
    float s = 0.0f;
#pragma unroll
    for (int i = 0; i < DDIM / 4; ++i) {
        float4 a = xr[i], c = mr[i];
        s += a.x * c.x + a.y * c.y + a.z * c.z + a.w * c.w;
    }
    outn[idx] = s;
}

// ---------------------------------------------------------------------------
// scratch zero + segment sum (atomic f32 adds into ws[1024 x 128])
// ---------------------------------------------------------------------------
__global__ void zero_kernel(float* __restrict__ ws, int n)
{
    int i = blockIdx.x * blockDim.x + threadIdx.x;
    if (i < n) ws[i] = 0.0f;
}

__global__ __launch_bounds__(128) void segsum_kernel(
    const float* __restrict__ x, const int* __restrict__ y,
    const int* __restrict__ vis, float* __restrict__ ws)
{
    int r = blockIdx.x;             // b*1024 + p
    int d = threadIdx.x;            // 0..127
    if (!vis[r]) return;
    int b = r >> 10, p = r & (N_POSI - 1);
    float v = x[(size_t)(b * XROW + p) * DDIM + d];
    atomicAdd(&ws[(size_t)y[r] * DDIM + d], v);
}

// ---------------------------------------------------------------------------
// y_idx passthrough (width decided at runtime from out_size partition)
// ---------------------------------------------------------------------------
__global__ void ycopy_kernel(const int* __restrict__ y, float* __restrict__ outy,
                             int y_elems)
{
    int i = blockIdx.x * blockDim.x + threadIdx.x;
    if (i >= BATCH * N_POSI) return;
    if (y_elems == 2 * BATCH * N_POSI)
        ((long long*)outy)[i] = (long long)y[i];
    else
        outy[i] = (float)y[i];
}

// ---------------------------------------------------------------------------
// new_memory: momentum blend (rows < 1024) / x_neg splice window / copy,
// then per-row L2 normalize. One 128-thread block (4 wave32) per row.
// ---------------------------------------------------------------------------
__global__ __launch_bounds__(128) void memnorm_kernel(
    const float* __restrict__ x, const float* __restrict__ mem,
    const float* __restrict__ ws, const int* __restrict__ lru,
    float* __restrict__ outm)
{
    const int r = blockIdx.x;       // 0..19999
    const int d = threadIdx.x;      // 0..127
    float v;
    if (r < N_POSI) {
        // mem_pos = mem*0.5 + (segsum/B)*0.5
        v = mem[(size_t)r * DDIM + d] * MOM
          + ws[(size_t)r * DDIM + d] * ((1.0f - MOM) / (float)BATCH);
    } else {
        const int start = N_POSI + lru[0] * N_NEGI * BATCH;
        if (r >= start && r < start + N_NEGI * BATCH) {
            int j = r - start;
            int b = j / N_NEGI, n = j % N_NEGI;
            v = x[(size_t)(b * XROW + N_POSI + n) * DDIM + d];
        } else {
            v = mem[(size_t)r * DDIM + d];
        }
    }
    // sum of squares: wave32 shuffle reduce, then 4 partials through LDS
    float s = v * v;
#pragma unroll
    for (int o = 16; o > 0; o >>= 1) s += __shfl_xor(s, o, 32);
    __shared__ float red[4];
    if ((threadIdx.x & 31) == 0) red[threadIdx.x >> 5] = s;
    __syncthreads();
    float tot   = red[0] + red[1] + red[2] + red[3];
    float denom = fmaxf(sqrtf(tot), 1e-12f);
    outm[(size_t)r * DDIM + d] = v / denom;
}

// ---------------------------------------------------------------------------
extern "C" void kernel_launch(void* const* d_in, const int* in_sizes, int n_in,
                              void* d_out, int out_size, void* d_ws, size_t ws_size,
                              hipStream_t stream)
{
    const float* x   = (const float*)d_in[0];   // (8, 1029, 128) f32
    const int*   y   = (const int*)  d_in[1];   // (8, 1024) int
    const int*   vis = (const int*)  d_in[2];   // (8, 1024) int
    const float* mem = (const float*)d_in[3];   // (20000, 128) f32
    const int*   lru = (const int*)  d_in[4];   // scalar int

    float* out = (float*)d_out;
    float* ws  = (float*)d_ws;

    const long long sim_elems   = (long long)BATCH * N_POSI * N_LEMI; // 163,840,000
    const long long noise_elems = (long long)BATCH * N_NEGI * N_POSI; // 40,960
    const long long mem_elems   = (long long)N_LEMI * DDIM;           // 2,560,000
    const long long y_elems     = (long long)out_size - sim_elems - noise_elems - mem_elems;

    float* out_sim   = out;
    float* out_y     = out_sim + sim_elems;
    float* out_noise = out_y + (y_elems > 0 ? y_elems : 0);
    float* out_mem   = out_noise + noise_elems;

    // 1) zero segment-sum scratch (deterministic per call)
    zero_kernel<<<(N_POSI * DDIM + 255) / 256, 256, 0, stream>>>(ws, N_POSI * DDIM);

    // 2) segment sum via global f32 atomics
    segsum_kernel<<<BATCH * N_POSI, DDIM, 0, stream>>>(x, y, vis, ws);

    // 3) main GEMM: 8192 x 20000, K=128, fp32 WMMA
    dim3 ggrid((N_LEMI + 63) / 64, (BATCH * N_POSI) / 64);   // (313, 128)
    sim_gemm_kernel<<<ggrid, 128, 0, stream>>>(x, mem, out_sim);

    // 4) noise similarity
    noise_kernel<<<(BATCH * N_NEGI * N_POSI + 255) / 256, 256, 0, stream>>>(x, mem, out_noise);

    // 5) y passthrough
    ycopy_kernel<<<(BATCH * N_POSI + 255) / 256, 256, 0, stream>>>(y, out_y, (int)y_elems);

    // 6) memory update + row L2 norm
    memnorm_kernel<<<N_LEMI, DDIM, 0, stream>>>(x, mem, ws, lru, out_mem);
}